// KubernetesA3TGCN_21827023798471
// MI455X (gfx1250) — compile-verified
//
#include <hip/hip_runtime.h>

#define N_NODES 50000
#define F_IN    8
#define HID     64
#define PERIODS 12
#define N_EDGES 1200000
#define FEAT    96            // F_IN * PERIODS
#define TILES   (N_NODES / 16)

typedef __attribute__((ext_vector_type(2))) float v2f;
typedef __attribute__((ext_vector_type(8))) float v8f;

// gfx1250-native transcendental tanh (V_TANH_F32). Trailing v_nop satisfies the
// TRANS-op result hazard (1 independent op / V_NOP before use).
static __device__ __forceinline__ float fast_tanh(float x) {
  float y;
  asm volatile("v_tanh_f32 %0, %1\n\tv_nop" : "=v"(y) : "v"(x));
  return y;
}
// branchless sigmoid via v_exp_f32 + v_rcp_f32
static __device__ __forceinline__ float fast_sigmoid(float x) {
  return __builtin_amdgcn_rcpf(1.0f + __builtin_amdgcn_exp2f(-1.44269504088896f * x));
}

// ---------------- degree / normalization ----------------
__global__ void k_init_deg(float* deg) {
  int i = blockIdx.x * blockDim.x + threadIdx.x;
  if (i < N_NODES) deg[i] = 1.0f;               // self-loop weight
}

__global__ void k_accum_deg(const long long* ei, const float* attr, float* deg) {
  int e = blockIdx.x * blockDim.x + threadIdx.x;
  if (e < N_EDGES) {
    int d = (int)ei[N_EDGES + e];
    atomicAdd(&deg[d], attr[e]);
  }
}

__global__ void k_rsqrt(float* deg) {
  int i = blockIdx.x * blockDim.x + threadIdx.x;
  if (i < N_NODES) deg[i] = rsqrtf(deg[i]);     // in-place -> dinv
}

// AX init with self-loop term: AX[i][:] = dinv[i]^2 * x[i][:]   (float4 lanes)
__global__ void k_init_ax(const float* __restrict__ x, const float* __restrict__ dinv,
                          float* __restrict__ AX) {
  int idx = blockIdx.x * blockDim.x + threadIdx.x;
  if (idx >= N_NODES * (FEAT / 4)) return;
  int i = idx / (FEAT / 4);
  int f = (idx - i * (FEAT / 4)) * 4;
  float dv = dinv[i];
  float s = dv * dv;
  float4 xv = *(const float4*)(x + i * FEAT + f);
  float4 r = make_float4(s * xv.x, s * xv.y, s * xv.z, s * xv.w);
  *(float4*)(AX + i * FEAT + f) = r;
}

// edge scatter: AX[dst][:] += dinv[src]*attr*dinv[dst] * x[src][:]  (float4 per thread)
__global__ void k_scatter(const float* __restrict__ x, const long long* __restrict__ ei,
                          const float* __restrict__ attr, const float* __restrict__ dinv,
                          float* __restrict__ AX) {
  int idx = blockIdx.x * blockDim.x + threadIdx.x;
  if (idx >= N_EDGES * (FEAT / 4)) return;
  int e = idx / (FEAT / 4);
  int f = (idx - e * (FEAT / 4)) * 4;
  int s = (int)ei[e];
  int d = (int)ei[N_EDGES + e];
  float w = dinv[s] * attr[e] * dinv[d];
  float4 xv = *(const float4*)(x + s * FEAT + f);
  float* base = AX + d * FEAT + f;
  atomicAdd(base + 0, w * xv.x);
  atomicAdd(base + 1, w * xv.y);
  atomicAdd(base + 2, w * xv.z);
  atomicAdd(base + 3, w * xv.w);
}

// ---------------- weight composition + softmax ----------------
// Wz_eff = Wcz(8x64) @ Wlz[:64](64x64); bz_eff = bcz @ Wlz[:64] + blz; same for h.
__global__ void k_prep(const float* att,
                       const float* Wcz, const float* bcz, const float* Wlz, const float* blz,
                       const float* Wch, const float* bch, const float* Wlh, const float* blh,
                       float* Wz, float* bz, float* Wh, float* bh, float* probs) {
  int j = threadIdx.x;                          // 64 threads
  for (int i = 0; i < F_IN; ++i) {
    float sz = 0.f, sh = 0.f;
    for (int k = 0; k < HID; ++k) {
      sz += Wcz[i * HID + k] * Wlz[k * HID + j];
      sh += Wch[i * HID + k] * Wlh[k * HID + j];
    }
    Wz[i * HID + j] = sz;
    Wh[i * HID + j] = sh;
  }
  float tz = blz[j], th = blh[j];
  for (int k = 0; k < HID; ++k) {
    tz += bcz[k] * Wlz[k * HID + j];
    th += bch[k] * Wlh[k * HID + j];
  }
  bz[j] = tz;
  bh[j] = th;
  if (j < PERIODS) {
    float m = -1e30f;
    for (int t = 0; t < PERIODS; ++t) m = fmaxf(m, att[t]);
    float s = 0.f;
    for (int t = 0; t < PERIODS; ++t) s += __expf(att[t] - m);
    probs[j] = __expf(att[j] - m) / s;
  }
}

// ---------------- node phase: WMMA f32 16x16x4 ----------------
__global__ __launch_bounds__(128) void k_node(
    const float* __restrict__ AX, const float* __restrict__ Wz, const float* __restrict__ bz,
    const float* __restrict__ Wh, const float* __restrict__ bh, const float* __restrict__ probs,
    const float* __restrict__ Wout, const float* __restrict__ bout, float* __restrict__ out) {
  __shared__ float smem[4][16][68];             // padded vs bank conflicts
  int wave = threadIdx.x >> 5;
  int lane = threadIdx.x & 31;
  int tile = blockIdx.x * 4 + wave;
  if (tile >= TILES) tile = TILES - 1;          // keep EXEC all-ones (dup write is benign)
  int m0 = tile * 16;
  int lo = lane & 15;
  int hi = lane >> 4;

  // B fragments: B[K][N], N on lanes; lanes>=16 hold K=2,3 (mirrors A layout)
  v2f Bz[2][4], Bh[2][4];
#pragma unroll
  for (int kc = 0; kc < 2; ++kc) {
    int kr = kc * 4 + hi * 2;
#pragma unroll
    for (int nt = 0; nt < 4; ++nt) {
      int col = nt * 16 + lo;
      Bz[kc][nt].x = Wz[kr * HID + col];
      Bz[kc][nt].y = Wz[(kr + 1) * HID + col];
      Bh[kc][nt].x = Wh[kr * HID + col];
      Bh[kc][nt].y = Wh[(kr + 1) * HID + col];
    }
  }
  v2f Bo[16];                                   // Wout (64x12) padded to 16 cols
#pragma unroll
  for (int kc = 0; kc < 16; ++kc) {
    int kr = kc * 4 + hi * 2;
    Bo[kc].x = (lo < PERIODS) ? Wout[kr * PERIODS + lo] : 0.f;
    Bo[kc].y = (lo < PERIODS) ? Wout[(kr + 1) * PERIODS + lo] : 0.f;
  }
  float bzv[4], bhv[4];
#pragma unroll
  for (int nt = 0; nt < 4; ++nt) { bzv[nt] = bz[nt * 16 + lo]; bhv[nt] = bh[nt * 16 + lo]; }

  v8f acc[4];
#pragma unroll
  for (int nt = 0; nt < 4; ++nt) acc[nt] = (v8f)0.0f;

  int rowbase = (m0 + lo) * FEAT;               // AX[n][f][t] at n*96 + f*12 + t
  for (int t = 0; t < PERIODS; ++t) {
    float p = probs[t];
    v2f A0, A1;                                 // A 16x4: M on lanes, K split lo/hi halves
    A0.x = AX[rowbase + (hi * 2 + 0) * PERIODS + t];
    A0.y = AX[rowbase + (hi * 2 + 1) * PERIODS + t];
    A1.x = AX[rowbase + (hi * 2 + 4) * PERIODS + t];
    A1.y = AX[rowbase + (hi * 2 + 5) * PERIODS + t];
#pragma unroll
    for (int nt = 0; nt < 4; ++nt) {
      v8f z = (v8f)0.0f, h = (v8f)0.0f;
      z = __builtin_amdgcn_wmma_f32_16x16x4_f32(false, A0, false, Bz[0][nt], (short)0, z, false, false);
      z = __builtin_amdgcn_wmma_f32_16x16x4_f32(false, A1, false, Bz[1][nt], (short)0, z, false, false);
      h = __builtin_amdgcn_wmma_f32_16x16x4_f32(false, A0, false, Bh[0][nt], (short)0, h, false, false);
      h = __builtin_amdgcn_wmma_f32_16x16x4_f32(false, A1, false, Bh[1][nt], (short)0, h, false, false);
#pragma unroll
      for (int v = 0; v < 8; ++v) {
        float zz = fast_sigmoid(z[v] + bzv[nt]);
        float hh = fast_tanh(h[v] + bhv[nt]);
        acc[nt][v] += p * (1.0f - zz) * hh;     // H0==0 -> H = (1-Z)*Ht
      }
    }
  }

  // relu + C-layout -> LDS -> A-layout for the readout matmul
#pragma unroll
  for (int nt = 0; nt < 4; ++nt)
#pragma unroll
    for (int v = 0; v < 8; ++v)
      smem[wave][v + hi * 8][nt * 16 + lo] = fmaxf(acc[nt][v], 0.0f);
  __syncthreads();

  v8f o = (v8f)0.0f;
#pragma unroll
  for (int kc = 0; kc < 16; ++kc) {
    v2f a;
    a.x = smem[wave][lo][kc * 4 + hi * 2];
    a.y = smem[wave][lo][kc * 4 + hi * 2 + 1];
    o = __builtin_amdgcn_wmma_f32_16x16x4_f32(false, a, false, Bo[kc], (short)0, o, false, false);
  }
  if (lo < PERIODS) {
    float bo = bout[lo];
#pragma unroll
    for (int v = 0; v < 8; ++v)
      out[(m0 + v + hi * 8) * PERIODS + lo] = o[v] + bo;
  }
}

// ---------------- launcher ----------------
extern "C" void kernel_launch(void* const* d_in, const int* in_sizes, int n_in,
                              void* d_out, int out_size, void* d_ws, size_t ws_size,
                              hipStream_t stream) {
  const float*     x    = (const float*)d_in[0];
  const long long* ei   = (const long long*)d_in[1];
  const float*     attr = (const float*)d_in[2];
  const float*     att  = (const float*)d_in[3];
  const float*     Wcz  = (const float*)d_in[4];
  const float*     bcz  = (const float*)d_in[5];
  // d_in[6], d_in[7] (Wcr, bcr) are dead: H0 == 0 kills the R path entirely.
  const float*     Wch  = (const float*)d_in[8];
  const float*     bch  = (const float*)d_in[9];
  const float*     Wlz  = (const float*)d_in[10];
  const float*     blz  = (const float*)d_in[11];
  // d_in[12], d_in[13] (Wlr, blr) dead too.
  const float*     Wlh  = (const float*)d_in[14];
  const float*     blh  = (const float*)d_in[15];
  const float*     Wout = (const float*)d_in[16];
  const float*     bout = (const float*)d_in[17];
  float* outp = (float*)d_out;

  float* ws    = (float*)d_ws;
  float* deg   = ws;                        // N
  float* AX    = deg + N_NODES;             // N*96
  float* Wz    = AX + (size_t)N_NODES * FEAT;   // 512
  float* bz    = Wz + F_IN * HID;           // 64
  float* Wh    = bz + HID;                  // 512
  float* bh    = Wh + F_IN * HID;           // 64
  float* probs = bh + HID;                  // 12

  k_init_deg<<<(N_NODES + 255) / 256, 256, 0, stream>>>(deg);
  k_accum_deg<<<(N_EDGES + 255) / 256, 256, 0, stream>>>(ei, attr, deg);
  k_rsqrt<<<(N_NODES + 255) / 256, 256, 0, stream>>>(deg);
  k_init_ax<<<(N_NODES * (FEAT / 4) + 255) / 256, 256, 0, stream>>>(x, deg, AX);
  k_scatter<<<(N_EDGES * (FEAT / 4) + 255) / 256, 256, 0, stream>>>(x, ei, attr, deg, AX);
  k_prep<<<1, 64, 0, stream>>>(att, Wcz, bcz, Wlz, blz, Wch, bch, Wlh, blh,
                               Wz, bz, Wh, bh, probs);
  k_node<<<(TILES + 3) / 4, 128, 0, stream>>>(AX, Wz, bz, Wh, bh, probs, Wout, bout, outp);
}